// MultiTaskElementNLLLoss_20469814132886
// MI455X (gfx1250) — compile-verified
//
#include <hip/hip_runtime.h>

// Custom clang ext-vectors so __builtin_nontemporal_load works (HIP float4 is a struct).
typedef float f32x2 __attribute__((ext_vector_type(2)));
typedef float f32x4 __attribute__((ext_vector_type(4)));
typedef float f32x8 __attribute__((ext_vector_type(8)));
typedef int   i32x4 __attribute__((ext_vector_type(4)));

#define TPB    256
#define BLOCKS 2048

// Branchless 3-way class pick (NUM_CLASS == 3): two v_cndmask ops.
__device__ __forceinline__ float pick3(float a, float b, float c, int t) {
    float r = (t == 1) ? b : a;
    return (t >= 2) ? c : r;
}

// Full wave32 sum via V_WMMA_F32_16X16X4_F32.
// A layout (16x4 f32): lanes 0-15 {vgpr0:K=0, vgpr1:K=1}, lanes 16-31 {vgpr0:K=2, vgpr1:K=3}.
// With a = (v, 0) per lane and B = ones: D[m][n] = v_m + v_{m+16} (replicated over n).
// Each lane holds 8 D rows -> local sum covers half the M range; xor-16 shuffle completes it.
// Requires EXEC all-ones: only called from uniform control flow.
__device__ __forceinline__ float wave_reduce32_wmma(float v) {
    f32x2 a; a.x = v;    a.y = 0.0f;
    f32x2 b; b.x = 1.0f; b.y = 1.0f;
    f32x8 c = {};
    f32x8 d = __builtin_amdgcn_wmma_f32_16x16x4_f32(
        /*neg_a=*/false, a, /*neg_b=*/false, b,
        /*c_mod=*/(short)0, c, /*reuse_a=*/false, /*reuse_b=*/false);
    float s = d[0] + d[1] + d[2] + d[3] + d[4] + d[5] + d[6] + d[7];
    s += __shfl_xor(s, 16, 32);
    return s;
}

__global__ __launch_bounds__(TPB) void nll_main_kernel(
    const f32x4* __restrict__ in0, const f32x4* __restrict__ in1, const f32x4* __restrict__ in2,
    const f32x4* __restrict__ cmap,
    const i32x4* __restrict__ tg0, const i32x4* __restrict__ tg1, const i32x4* __restrict__ tg2,
    float* __restrict__ partials, int nquads) {
    const int stride = gridDim.x * blockDim.x;
    float acc = 0.0f;

    for (int q = blockIdx.x * blockDim.x + threadIdx.x; q < nquads; q += stride) {
        // 4 rows per quad: streamed once, dataset > L2 -> non-temporal.
        f32x4 cm = __builtin_nontemporal_load(&cmap[q]);
        i32x4 a0 = __builtin_nontemporal_load(&tg0[q]);
        i32x4 a1 = __builtin_nontemporal_load(&tg1[q]);
        i32x4 a2 = __builtin_nontemporal_load(&tg2[q]);

        f32x4 u0 = __builtin_nontemporal_load(&in0[3 * q + 0]);
        f32x4 u1 = __builtin_nontemporal_load(&in0[3 * q + 1]);
        f32x4 u2 = __builtin_nontemporal_load(&in0[3 * q + 2]);
        f32x4 v0 = __builtin_nontemporal_load(&in1[3 * q + 0]);
        f32x4 v1 = __builtin_nontemporal_load(&in1[3 * q + 1]);
        f32x4 v2 = __builtin_nontemporal_load(&in1[3 * q + 2]);
        f32x4 w0 = __builtin_nontemporal_load(&in2[3 * q + 0]);
        f32x4 w1 = __builtin_nontemporal_load(&in2[3 * q + 1]);
        f32x4 w2 = __builtin_nontemporal_load(&in2[3 * q + 2]);

        // row r of a quad lives at floats [3r, 3r+2] of the 12-float span.
        float r0 =         pick3(u0.x, u0.y, u0.z, a0.x)
                 + 0.50f * pick3(v0.x, v0.y, v0.z, a1.x)
                 + 0.25f * pick3(w0.x, w0.y, w0.z, a2.x);
        float r1 =         pick3(u0.w, u1.x, u1.y, a0.y)
                 + 0.50f * pick3(v0.w, v1.x, v1.y, a1.y)
                 + 0.25f * pick3(w0.w, w1.x, w1.y, a2.y);
        float r2 =         pick3(u1.z, u1.w, u2.x, a0.z)
                 + 0.50f * pick3(v1.z, v1.w, v2.x, a1.z)
                 + 0.25f * pick3(w1.z, w1.w, w2.x, a2.z);
        float r3 =         pick3(u2.y, u2.z, u2.w, a0.w)
                 + 0.50f * pick3(v2.y, v2.z, v2.w, a1.w)
                 + 0.25f * pick3(w2.y, w2.z, w2.w, a2.w);

        acc = fmaf(cm.x, r0, acc);
        acc = fmaf(cm.y, r1, acc);
        acc = fmaf(cm.z, r2, acc);
        acc = fmaf(cm.w, r3, acc);
    }

    // Uniform trip counts -> EXEC is all-ones here.
    float wsum = wave_reduce32_wmma(acc);

    __shared__ float sm[TPB / 32];
    const int lane = threadIdx.x & 31;
    const int wv   = threadIdx.x >> 5;
    if (lane == 0) sm[wv] = wsum;
    __syncthreads();
    if (threadIdx.x == 0) {
        float t = 0.0f;
#pragma unroll
        for (int i = 0; i < TPB / 32; ++i) t += sm[i];
        partials[blockIdx.x] = t;
    }
}

__global__ __launch_bounds__(TPB) void nll_final_kernel(
    const float* __restrict__ partials, float* __restrict__ out, int n, float scale) {
    float acc = 0.0f;
    for (int i = threadIdx.x; i < n; i += TPB) acc += partials[i];  // fixed order: deterministic

    float wsum = wave_reduce32_wmma(acc);

    __shared__ float sm[TPB / 32];
    const int lane = threadIdx.x & 31;
    const int wv   = threadIdx.x >> 5;
    if (lane == 0) sm[wv] = wsum;
    __syncthreads();
    if (threadIdx.x == 0) {
        float t = 0.0f;
#pragma unroll
        for (int i = 0; i < TPB / 32; ++i) t += sm[i];
        out[0] = t * scale;  // fold -(1/N) and final negate into one multiply
    }
}

extern "C" void kernel_launch(void* const* d_in, const int* in_sizes, int n_in,
                              void* d_out, int out_size, void* d_ws, size_t ws_size,
                              hipStream_t stream) {
    const f32x4* in0  = (const f32x4*)d_in[0];
    const f32x4* in1  = (const f32x4*)d_in[1];
    const f32x4* in2  = (const f32x4*)d_in[2];
    const f32x4* cmap = (const f32x4*)d_in[3];
    const i32x4* tg0  = (const i32x4*)d_in[4];
    const i32x4* tg1  = (const i32x4*)d_in[5];
    const i32x4* tg2  = (const i32x4*)d_in[6];

    float* partials = (float*)d_ws;            // BLOCKS floats = 8 KB scratch
    const int n      = in_sizes[3];            // N rows (cmap length)
    const int nquads = n >> 2;                 // 4 rows per thread-iteration
    const float scale = -1.0f / (float)n;

    nll_main_kernel<<<BLOCKS, TPB, 0, stream>>>(in0, in1, in2, cmap, tg0, tg1, tg2,
                                                partials, nquads);
    nll_final_kernel<<<1, TPB, 0, stream>>>(partials, (float*)d_out, BLOCKS, scale);
}